// JobGCN_2843268349978
// MI455X (gfx1250) — compile-verified
//
#include <hip/hip_runtime.h>
#include <cstdint>

typedef __attribute__((ext_vector_type(2))) float v2f;
typedef __attribute__((ext_vector_type(8))) float v8f;

#define WAVES 8
#define TPB   (WAVES * 32)

// ---------------- utility ----------------
__global__ void k_zero_u32(uint32_t* __restrict__ p, size_t n) {
  size_t i = (size_t)blockIdx.x * blockDim.x + threadIdx.x;
  size_t st = (size_t)gridDim.x * blockDim.x;
  for (; i < n; i += st) p[i] = 0u;
}

// ---------------- degrees (exact integer counts) ----------------
__global__ void k_degrees(const int* __restrict__ src, const int* __restrict__ dst,
                          int n_edge, uint32_t* __restrict__ deg_var,
                          uint32_t* __restrict__ deg_con) {
  int e = blockIdx.x * blockDim.x + threadIdx.x;
  if (e < n_edge) {
    atomicAdd(&deg_var[src[e]], 1u);
    atomicAdd(&deg_con[dst[e]], 1u);
  }
}

// ---------------- var embedding: relu([c,x]@Wv+bv) * deg^-1/2 ----------------
__global__ void k_embed(const float* __restrict__ var_c, const float* __restrict__ var_x,
                        const float* __restrict__ Wv, const float* __restrict__ bv,
                        const uint32_t* __restrict__ deg_var,
                        float* __restrict__ hv, int n_var, int H) {
  int i = blockIdx.x * blockDim.x + threadIdx.x;
  if (i >= n_var) return;
  float c = var_c[i], x = var_x[i];
  float s = rsqrtf(fmaxf((float)deg_var[i], 1.0f));
  float* out = hv + (size_t)i * H;
  for (int f = 0; f < H; ++f) {
    float v = fmaf(c, Wv[f], fmaf(x, Wv[H + f], bv[f]));
    out[f] = fmaxf(v, 0.0f) * s;
  }
}

// ---------------- edge gather/scatter: agg[dst] += h[src]*w ----------------
// H==10 fast path: 40B rows are 8B aligned -> gather with 5x global_load_b64.
__global__ void k_edge_scatter10(const float* __restrict__ hsrc, const int* __restrict__ isrc,
                                 const int* __restrict__ idst, const float* __restrict__ w,
                                 int n_edge, float* __restrict__ agg) {
  int e = blockIdx.x * blockDim.x + threadIdx.x;
  if (e >= n_edge) return;
  int s = isrc[e], d = idst[e];
  float wt = w[e];
  const float2* hr = (const float2*)(hsrc + (size_t)s * 10);
  float* ar = agg + (size_t)d * 10;
#pragma unroll
  for (int p = 0; p < 5; ++p) {
    float2 v = hr[p];
    unsafeAtomicAdd(&ar[2 * p + 0], v.x * wt);  // global_atomic_add_f32, no return
    unsafeAtomicAdd(&ar[2 * p + 1], v.y * wt);
  }
}

__global__ void k_edge_scatter(const float* __restrict__ hsrc, const int* __restrict__ isrc,
                               const int* __restrict__ idst, const float* __restrict__ w,
                               int n_edge, int H, float* __restrict__ agg) {
  int e = blockIdx.x * blockDim.x + threadIdx.x;
  if (e >= n_edge) return;
  int s = isrc[e], d = idst[e];
  float wt = w[e];
  const float* hr = hsrc + (size_t)s * H;
  float* ar = agg + (size_t)d * H;
  for (int f = 0; f < H; ++f)
    unsafeAtomicAdd(&ar[f], hr[f] * wt);
}

// ---------------- WMMA helper: one 16x16 f32 layer over a tile in LDS ----------------
__device__ __forceinline__ v8f wmma_layer(const float (*t)[17], int lm, int hi,
                                          const v2f wr[4], float bias) {
  v8f acc;
#pragma unroll
  for (int j = 0; j < 8; ++j) acc[j] = bias;
#pragma unroll
  for (int c = 0; c < 4; ++c) {
    v2f A;
    A[0] = t[lm][4 * c + 2 * hi];
    A[1] = t[lm][4 * c + 1 + 2 * hi];
    acc = __builtin_amdgcn_wmma_f32_16x16x4_f32(false, A, false, wr[c],
                                                (short)0, acc, false, false);
  }
  return acc;
}

// ---------------- con side: hc = relu((agg*dc^-1/2)@W2+b2) * dc^-1/2 ----------------
__global__ void k_con_wmma(const float* __restrict__ agg_con, const uint32_t* __restrict__ deg_con,
                           const float* __restrict__ W2, const float* __restrict__ b2,
                           float* __restrict__ hc, int n_con, int H) {
  __shared__ float wpad[16][16];
  __shared__ float bpad[16];
  __shared__ float tile[WAVES][16][17];
  __shared__ float rsc[WAVES][16];

  for (int idx = threadIdx.x; idx < 256; idx += blockDim.x) {
    int r = idx >> 4, cc = idx & 15;
    wpad[r][cc] = (r < H && cc < H) ? W2[r * H + cc] : 0.0f;
    if (r == 0) bpad[cc] = (cc < H) ? b2[cc] : 0.0f;
  }
  __syncthreads();

  int lane = threadIdx.x & 31, wv = threadIdx.x >> 5;
  int lm = lane & 15, hi = lane >> 4;

  v2f wr[4];
#pragma unroll
  for (int c = 0; c < 4; ++c) {
    wr[c][0] = wpad[4 * c + 2 * hi][lm];
    wr[c][1] = wpad[4 * c + 1 + 2 * hi][lm];
  }
  float br = bpad[lm];

  int base = (blockIdx.x * WAVES + wv) * 16;
  if (base >= n_con) return;

  for (int idx = lane; idx < 256; idx += 32) {
    int r = idx >> 4, col = idx & 15;
    int node = base + r;
    float v = 0.0f;
    if (node < n_con && col < H) {
      float s = rsqrtf(fmaxf((float)deg_con[node], 1.0f));
      v = agg_con[(size_t)node * H + col] * s;
    }
    tile[wv][r][col] = v;
  }
  if (lane < 16) {
    int node = base + lane;
    rsc[wv][lane] = (node < n_con) ? rsqrtf(fmaxf((float)deg_con[node], 1.0f)) : 0.0f;
  }
  __builtin_amdgcn_wave_barrier();  // same-wave DS ops are in-order

  v8f acc = wmma_layer(tile[wv], lm, hi, wr, br);

  // stage relu(acc)*rsc back to LDS, then store row-major coalesced
#pragma unroll
  for (int j = 0; j < 8; ++j)
    tile[wv][j + 8 * hi][lm] = fmaxf(acc[j], 0.0f) * rsc[wv][j + 8 * hi];
  __builtin_amdgcn_wave_barrier();

  int nelem = 16 * H;
  for (int idx = lane; idx < nelem; idx += 32) {
    int r = idx / H, col = idx % H;
    int node = base + r;
    if (node < n_con) hc[(size_t)node * H + col] = tile[wv][r][col];
  }
}

// ---------------- var side: 4-layer MLP per node + deterministic logit reduction ------
__global__ void k_var_mlp(const float* __restrict__ agg_var, const uint32_t* __restrict__ deg_var,
                          const float* __restrict__ W2,  const float* __restrict__ b2,
                          const float* __restrict__ Wo1, const float* __restrict__ bo1,
                          const float* __restrict__ Wo2, const float* __restrict__ bo2,
                          const float* __restrict__ Wo3, const float* __restrict__ bo3,
                          float* __restrict__ partials, int n_var, int H) {
  __shared__ float wpad[4][16][16];
  __shared__ float bpad[4][16];
  __shared__ float tile[WAVES][16][17];
  __shared__ float wsum[WAVES];

  for (int idx = threadIdx.x; idx < 256; idx += blockDim.x) {
    int r = idx >> 4, cc = idx & 15;
    wpad[0][r][cc] = (r < H && cc < H) ? W2[r * H + cc] : 0.0f;
    wpad[1][r][cc] = (r < H && cc < H) ? Wo1[r * H + cc] : 0.0f;
    wpad[2][r][cc] = (r < H && cc < H) ? Wo2[r * H + cc] : 0.0f;
    wpad[3][r][cc] = (r < H && cc == 0) ? Wo3[r] : 0.0f;
    if (r == 0) {
      bpad[0][cc] = (cc < H) ? b2[cc] : 0.0f;
      bpad[1][cc] = (cc < H) ? bo1[cc] : 0.0f;
      bpad[2][cc] = (cc < H) ? bo2[cc] : 0.0f;
      bpad[3][cc] = (cc == 0) ? bo3[0] : 0.0f;
    }
  }
  __syncthreads();

  int lane = threadIdx.x & 31, wv = threadIdx.x >> 5;
  int lm = lane & 15, hi = lane >> 4;

  v2f wr[4][4];
  float br[4];
#pragma unroll
  for (int l = 0; l < 4; ++l) {
#pragma unroll
    for (int c = 0; c < 4; ++c) {
      wr[l][c][0] = wpad[l][4 * c + 2 * hi][lm];
      wr[l][c][1] = wpad[l][4 * c + 1 + 2 * hi][lm];
    }
    br[l] = bpad[l][lm];
  }

  float tile_sum = 0.0f;
  int base = (blockIdx.x * WAVES + wv) * 16;
  if (base < n_var) {
    for (int idx = lane; idx < 256; idx += 32) {
      int r = idx >> 4, col = idx & 15;
      int node = base + r;
      float v = 0.0f;
      if (node < n_var && col < H) {
        float s = rsqrtf(fmaxf((float)deg_var[node], 1.0f));
        v = agg_var[(size_t)node * H + col] * s;
      }
      tile[wv][r][col] = v;
    }
    __builtin_amdgcn_wave_barrier();

    v8f acc;
#pragma unroll
    for (int l = 0; l < 4; ++l) {
      acc = wmma_layer(tile[wv], lm, hi, wr[l], br[l]);
      if (l < 3) {
#pragma unroll
        for (int j = 0; j < 8; ++j) tile[wv][j + 8 * hi][lm] = fmaxf(acc[j], 0.0f);
        __builtin_amdgcn_wave_barrier();
      }
    }
    // logits live in feat 0 == lanes 0 and 16; D: VGPR j lane l = node (j+8*hi)
    float t = 0.0f;
#pragma unroll
    for (int j = 0; j < 8; ++j) {
      int node = base + j + 8 * hi;
      if (node < n_var) t += acc[j];
    }
    tile_sum = __shfl(t, 0, 32) + __shfl(t, 16, 32);
  }

  if (lane == 0) wsum[wv] = tile_sum;
  __syncthreads();
  if (threadIdx.x == 0) {
    float s = 0.0f;
#pragma unroll
    for (int w = 0; w < WAVES; ++w) s += wsum[w];
    partials[blockIdx.x] = s;  // fixed order -> deterministic
  }
}

__global__ void k_finalize(const float* __restrict__ partials, int n, float inv_n,
                           float* __restrict__ out) {
  __shared__ float sh[256];
  float s = 0.0f;
  for (int i = threadIdx.x; i < n; i += 256) s += partials[i];
  sh[threadIdx.x] = s;
  __syncthreads();
  for (int off = 128; off > 0; off >>= 1) {
    if ((int)threadIdx.x < off) sh[threadIdx.x] += sh[threadIdx.x + off];
    __syncthreads();
  }
  if (threadIdx.x == 0) out[0] = sh[0] * inv_n;
}

// ---------------- driver ----------------
extern "C" void kernel_launch(void* const* d_in, const int* in_sizes, int n_in,
                              void* d_out, int out_size, void* d_ws, size_t ws_size,
                              hipStream_t stream) {
  const float* var_c   = (const float*)d_in[0];
  const float* var_x   = (const float*)d_in[1];
  const int*   edge_src= (const int*)d_in[3];
  const int*   edge_dst= (const int*)d_in[4];
  const float* edge_w  = (const float*)d_in[5];
  const float* Wv  = (const float*)d_in[6];
  const float* bv  = (const float*)d_in[7];
  // d_in[8..11] (Wc,bc,W1,b1) are dead in the reference
  const float* W2  = (const float*)d_in[12];
  const float* b2  = (const float*)d_in[13];
  const float* Wo1 = (const float*)d_in[14];
  const float* bo1 = (const float*)d_in[15];
  const float* Wo2 = (const float*)d_in[16];
  const float* bo2 = (const float*)d_in[17];
  const float* Wo3 = (const float*)d_in[18];
  const float* bo3 = (const float*)d_in[19];

  const int n_var  = in_sizes[0];
  const int n_con  = in_sizes[2];
  const int n_edge = in_sizes[3];
  const int H      = in_sizes[7];  // bv has H elements (H <= 16 required)

  // workspace carving (256B aligned); agg_var aliases hv (hv is dead after pass 1)
  char* ws = (char*)d_ws;
  size_t off = 0;
  auto carve = [&](size_t bytes) {
    size_t r = off;
    off += (bytes + 255) & ~(size_t)255;
    return r;
  };
  uint32_t* deg_var = (uint32_t*)(ws + carve((size_t)n_var * 4));
  uint32_t* deg_con = (uint32_t*)(ws + carve((size_t)n_con * 4));
  float* hv      = (float*)(ws + carve((size_t)n_var * H * 4));
  float* agg_con = (float*)(ws + carve((size_t)n_con * H * 4));
  float* hc      = (float*)(ws + carve((size_t)n_con * H * 4));
  const int tiles_var = (n_var + 15) / 16;
  const int tiles_con = (n_con + 15) / 16;
  const int blocks_mlp = (tiles_var + WAVES - 1) / WAVES;
  const int blocks_con = (tiles_con + WAVES - 1) / WAVES;
  float* partials = (float*)(ws + carve((size_t)blocks_mlp * 4));
  float* agg_var = hv;

  auto zgrid = [](size_t n) {
    size_t b = (n + 255) / 256;
    if (b > 8192) b = 8192;
    if (b < 1) b = 1;
    return (int)b;
  };
  const int eblocks = (n_edge + 255) / 256;
  const int vblocks = (n_var + 255) / 256;

  k_zero_u32<<<zgrid(n_var), 256, 0, stream>>>(deg_var, (size_t)n_var);
  k_zero_u32<<<zgrid(n_con), 256, 0, stream>>>(deg_con, (size_t)n_con);
  k_zero_u32<<<zgrid((size_t)n_con * H), 256, 0, stream>>>((uint32_t*)agg_con, (size_t)n_con * H);

  k_degrees<<<eblocks, 256, 0, stream>>>(edge_src, edge_dst, n_edge, deg_var, deg_con);
  k_embed<<<vblocks, 256, 0, stream>>>(var_c, var_x, Wv, bv, deg_var, hv, n_var, H);

  // v->c pass
  if (H == 10)
    k_edge_scatter10<<<eblocks, 256, 0, stream>>>(hv, edge_src, edge_dst, edge_w, n_edge, agg_con);
  else
    k_edge_scatter<<<eblocks, 256, 0, stream>>>(hv, edge_src, edge_dst, edge_w, n_edge, H, agg_con);
  k_con_wmma<<<blocks_con, TPB, 0, stream>>>(agg_con, deg_con, W2, b2, hc, n_con, H);

  // c->v pass (agg_var aliases hv, re-zeroed now that hv is dead)
  k_zero_u32<<<zgrid((size_t)n_var * H), 256, 0, stream>>>((uint32_t*)agg_var, (size_t)n_var * H);
  if (H == 10)
    k_edge_scatter10<<<eblocks, 256, 0, stream>>>(hc, edge_dst, edge_src, edge_w, n_edge, agg_var);
  else
    k_edge_scatter<<<eblocks, 256, 0, stream>>>(hc, edge_dst, edge_src, edge_w, n_edge, H, agg_var);

  k_var_mlp<<<blocks_mlp, TPB, 0, stream>>>(agg_var, deg_var, W2, b2, Wo1, bo1, Wo2, bo2,
                                            Wo3, bo3, partials, n_var, H);
  k_finalize<<<1, 256, 0, stream>>>(partials, blocks_mlp, 1.0f / (float)n_var, (float*)d_out);
}